// ViT_Branch_14474039787788
// MI455X (gfx1250) — compile-verified
//
#include <hip/hip_runtime.h>
#include <hip/hip_bf16.h>
#include <math.h>

typedef float v2f __attribute__((ext_vector_type(2)));
typedef float v8f __attribute__((ext_vector_type(8)));

#define C_DIM 768
#define LDS_STRIDE 772          // 768 + 4 pad: lane l -> bank (4l + k) % 64, conflict-free
#define NFC_PAD 208             // 200 classes padded to 13*16
#define BS 16

// ---------------------------------------------------------------------------
// K0: build weight buffers in WMMA B-fragment pair layout + concatenated bias.
// Pair layout: Wtp[(k0/4)*2 + half][n] = float2{ W[n][k0+2h], W[n][k0+2h+1] }
// -> one coalesced global_load_b64 per WMMA per lane in the GEMM loops.
//  Wtp_fc    : (768/2) x 208 float2   (n>=200 zero)
//  Wtp_parts : (768/2) x 768 float2   (cols 0..191=W_l, 192..383=W_m, rest=W_s)
//  bias_cat  : 768
// ---------------------------------------------------------------------------
__global__ void prep_kernel(const float* __restrict__ W_fc,
                            const float* __restrict__ W_l,
                            const float* __restrict__ W_m,
                            const float* __restrict__ W_s,
                            const float* __restrict__ b_l,
                            const float* __restrict__ b_m,
                            const float* __restrict__ b_s,
                            float2* __restrict__ Wtp_fc,
                            float2* __restrict__ Wtp_parts,
                            float* __restrict__ bias_cat) {
  const int P0 = (C_DIM / 2) * NFC_PAD;   // float2 count
  const int P1 = (C_DIM / 2) * C_DIM;
  int gid = blockIdx.x * blockDim.x + threadIdx.x;
  if (gid < P0) {
    int n = gid % NFC_PAD, kp = gid / NFC_PAD;
    int k = (kp >> 1) * 4 + (kp & 1) * 2;           // k0 + 2*half
    float2 v = {0.0f, 0.0f};
    if (n < 200) {
      v.x = W_fc[(size_t)n * C_DIM + k];
      v.y = W_fc[(size_t)n * C_DIM + k + 1];
    }
    Wtp_fc[gid] = v;
  } else if (gid < P0 + P1) {
    int idx = gid - P0;
    int n = idx % C_DIM, kp = idx / C_DIM;
    int k = (kp >> 1) * 4 + (kp & 1) * 2;
    const float* W; int nn;
    if (n < 192)      { W = W_l; nn = n; }
    else if (n < 384) { W = W_m; nn = n - 192; }
    else              { W = W_s; nn = n - 384; }
    float2 v;
    v.x = W[(size_t)nn * C_DIM + k];
    v.y = W[(size_t)nn * C_DIM + k + 1];
    Wtp_parts[idx] = v;
  } else if (gid < P0 + P1 + C_DIM) {
    int n = gid - P0 - P1;
    bias_cat[n] = (n < 192) ? b_l[n] : (n < 384 ? b_m[n - 192] : b_s[n - 384]);
  }
}

// ---------------------------------------------------------------------------
// K1: logits = x @ W_fc^T + b_fc ; ml[b,t] = 1 / sum(exp(logit - max_logit))
// One wave per 16-row tile. WMMA f32 16x16x4. Per-lane online softmax state
// (each lane only ever owns column n = nb*16 + col), merged once at the end
// with a stabilized 4-round xor-shuffle log-sum-exp combine.
// ---------------------------------------------------------------------------
__global__ __launch_bounds__(32) void cls_softmax_kernel(
    const float* __restrict__ x, const float2* __restrict__ Wtp_fc,
    const float* __restrict__ b_fc, float* __restrict__ ml) {
  __shared__ float lds[16 * LDS_STRIDE];
  const int tile = blockIdx.x;            // 0..783
  const int b = tile / 49;
  const int t0 = (tile % 49) * 16;
  const int lane = threadIdx.x;
  const float* xb = x + ((size_t)b * 784 + t0) * C_DIM;

  // Stage 16 x 768 x-tile into LDS (coalesced float4 rows).
  for (int idx = lane; idx < 16 * (C_DIM / 4); idx += 32) {
    int r = idx / (C_DIM / 4);
    int c4 = (idx % (C_DIM / 4)) * 4;
    float4 v = *(const float4*)(xb + (size_t)r * C_DIM + c4);
    *(float4*)&lds[r * LDS_STRIDE + c4] = v;
  }
  __syncthreads();

  const int half = lane >> 4, col = lane & 15, koff = half * 2;
  float mrun[8], srun[8];
  #pragma unroll
  for (int v = 0; v < 8; ++v) { mrun[v] = -__builtin_inff(); srun[v] = 0.0f; }

  for (int nb = 0; nb < 13; ++nb) {
    const int n = nb * 16 + col;
    v8f acc = {0.f, 0.f, 0.f, 0.f, 0.f, 0.f, 0.f, 0.f};
    #pragma unroll 8
    for (int k0 = 0; k0 < C_DIM; k0 += 4) {
      v2f a = *(const v2f*)&lds[col * LDS_STRIDE + k0 + koff];
      float2 bp = Wtp_fc[(size_t)((k0 >> 2) * 2 + half) * NFC_PAD + n];
      v2f bb = {bp.x, bp.y};
      acc = __builtin_amdgcn_wmma_f32_16x16x4_f32(false, a, false, bb,
                                                  (short)0, acc, false, false);
    }
    if (n < 200) {                           // per-lane online update, no shuffles
      const float bias = b_fc[n];
      #pragma unroll
      for (int v = 0; v < 8; ++v) {
        float cv = acc[v] + bias;
        float mnew = fmaxf(mrun[v], cv);
        srun[v] = srun[v] * expf(mrun[v] - mnew) + expf(cv - mnew);
        mrun[v] = mnew;
      }
    }
  }

  // Merge (m, s) across the 16-lane half (xor masks < 16 stay in the half).
  #pragma unroll
  for (int v = 0; v < 8; ++v) {
    float m = mrun[v], s = srun[v];
    #pragma unroll
    for (int msk = 1; msk < 16; msk <<= 1) {
      float mo = __shfl_xor(m, msk, 32);
      float so = __shfl_xor(s, msk, 32);
      float mn = fmaxf(m, mo);
      s = s * expf(m - mn) + so * expf(mo - mn);
      m = mn;
    }
    srun[v] = s;
  }
  if (col == 0) {
    #pragma unroll
    for (int v = 0; v < 8; ++v) {
      int row = v + half * 8;                     // C/D layout: M = v + 8*half
      ml[(size_t)b * 784 + t0 + row] = 1.0f / srun[v];
    }
  }
}

// ---------------------------------------------------------------------------
// K2: part_logits = mask * mean(8x8 window, pad 2, stride 3) ; sel = argmax
// ---------------------------------------------------------------------------
__global__ void select_kernel(const float* __restrict__ ml,
                              const float* __restrict__ mask,
                              int* __restrict__ sel) {
  __shared__ float pl[81];
  const int b = blockIdx.x, tid = threadIdx.x;
  if (tid < 81) {
    int pi = tid / 9, pj = tid % 9;
    float s = 0.0f;
    for (int i = 0; i < 8; ++i) {
      int h = pi * 3 + i - 2;
      if (h < 0 || h >= 28) continue;
      for (int j = 0; j < 8; ++j) {
        int w = pj * 3 + j - 2;
        if (w < 0 || w >= 28) continue;
        s += ml[(size_t)b * 784 + h * 28 + w];
      }
    }
    pl[tid] = mask[b * 81 + tid] * (s * (1.0f / 64.0f));
  }
  __syncthreads();
  if (tid == 0) {
    float best = pl[0]; int bi = 0;
    for (int p = 1; p < 81; ++p)
      if (pl[p] > best) { best = pl[p]; bi = p; }   // first-max like jnp.argmax
    sel[b] = bi;
  }
}

// ---------------------------------------------------------------------------
// K3: fused gather + bilinear staging of outl/outm/outs 16-row tiles into LDS
// (each A-row is a <=4-term weighted sum of contiguous x rows), then WMMA f32
// 16x16x4 GEMM vs Wtp_parts; writes ff_cat slice + bias.
// grid: 192 one-wave blocks = b(16) x mtile(4) x part(3).
// ---------------------------------------------------------------------------
__global__ __launch_bounds__(32) void parts_gemm_kernel(
    const float* __restrict__ x, const int* __restrict__ sel,
    const float2* __restrict__ Wtp_parts, const float* __restrict__ bias_cat,
    float* __restrict__ out) {
  __shared__ float lds[16 * LDS_STRIDE];
  const int blk = blockIdx.x;
  const int b = blk / 12;
  const int rem = blk % 12;
  const int mtile = rem / 3;
  const int part = rem % 3;                 // 0=l(8x8), 1=m(6x6->8x8), 2=s(4x4->8x8)
  const int lane = threadIdx.x;
  const int s = sel[b];
  const int pi = s / 9, pj = s % 9;
  const float* xb = x + (size_t)b * 784 * C_DIM;

  for (int lr = 0; lr < 16; ++lr) {
    const int r = mtile * 16 + lr;          // output spatial index y*8+x
    const int yy = r >> 3, xx = r & 7;
    int   tt[4];
    float ww[4];
    if (part == 0) {
      int h = pi * 3 + yy - 2, w = pj * 3 + xx - 2;
      bool ok = (h >= 0 && h < 28 && w >= 0 && w < 28);
      tt[0] = ok ? h * 28 + w : 0; ww[0] = ok ? 1.0f : 0.0f;
      tt[1] = tt[2] = tt[3] = 0;  ww[1] = ww[2] = ww[3] = 0.0f;
    } else {
      const float scale = (part == 1) ? 0.75f : 0.5f;  // 6/8 or 4/8
      const int inmax   = (part == 1) ? 5 : 3;
      const int pad     = (part == 1) ? 1 : 0;
      float sy = (yy + 0.5f) * scale - 0.5f;
      float sx = (xx + 0.5f) * scale - 0.5f;
      int iy0 = (int)floorf(sy); float fy = sy - (float)iy0;
      int ix0 = (int)floorf(sx); float fx = sx - (float)ix0;
      int iy[2] = { min(max(iy0, 0), inmax), min(max(iy0 + 1, 0), inmax) };
      int ix[2] = { min(max(ix0, 0), inmax), min(max(ix0 + 1, 0), inmax) };
      float wy[2] = { 1.0f - fy, fy };
      float wx[2] = { 1.0f - fx, fx };
      #pragma unroll
      for (int q = 0; q < 4; ++q) {
        int h = pi * 3 + iy[q >> 1] - pad;
        int w = pj * 3 + ix[q & 1] - pad;
        bool ok = (h >= 0 && h < 28 && w >= 0 && w < 28);  // unfold zero-pad
        tt[q] = ok ? h * 28 + w : 0;
        ww[q] = ok ? wy[q >> 1] * wx[q & 1] : 0.0f;
      }
    }
    const float4* r0 = (const float4*)(xb + (size_t)tt[0] * C_DIM);
    const float4* r1 = (const float4*)(xb + (size_t)tt[1] * C_DIM);
    const float4* r2 = (const float4*)(xb + (size_t)tt[2] * C_DIM);
    const float4* r3 = (const float4*)(xb + (size_t)tt[3] * C_DIM);
    for (int c4 = lane; c4 < C_DIM / 4; c4 += 32) {
      float4 v0 = r0[c4], v1 = r1[c4], v2 = r2[c4], v3 = r3[c4];
      float4 o;
      o.x = ww[0] * v0.x + ww[1] * v1.x + ww[2] * v2.x + ww[3] * v3.x;
      o.y = ww[0] * v0.y + ww[1] * v1.y + ww[2] * v2.y + ww[3] * v3.y;
      o.z = ww[0] * v0.z + ww[1] * v1.z + ww[2] * v2.z + ww[3] * v3.z;
      o.w = ww[0] * v0.w + ww[1] * v1.w + ww[2] * v2.w + ww[3] * v3.w;
      *(float4*)&lds[lr * LDS_STRIDE + c4 * 4] = o;
    }
  }
  __syncthreads();

  const int half = lane >> 4, col = lane & 15, koff = half * 2;
  const int nbStart = (part == 0) ? 0 : (part == 1 ? 12 : 24);
  const int nbCnt   = (part == 2) ? 24 : 12;
  float* outp = out + ((size_t)b * 64 + mtile * 16) * C_DIM;

  for (int q = 0; q < nbCnt; ++q) {
    const int n = (nbStart + q) * 16 + col;
    v8f acc = {0.f, 0.f, 0.f, 0.f, 0.f, 0.f, 0.f, 0.f};
    #pragma unroll 8
    for (int k0 = 0; k0 < C_DIM; k0 += 4) {
      v2f a = *(const v2f*)&lds[col * LDS_STRIDE + k0 + koff];
      float2 bp = Wtp_parts[(size_t)((k0 >> 2) * 2 + half) * C_DIM + n];
      v2f bb = {bp.x, bp.y};
      acc = __builtin_amdgcn_wmma_f32_16x16x4_f32(false, a, false, bb,
                                                  (short)0, acc, false, false);
    }
    const float bias = bias_cat[n];
    #pragma unroll
    for (int v = 0; v < 8; ++v) {
      int M = v + half * 8;
      outp[(size_t)M * C_DIM + n] = acc[v] + bias;
    }
  }
}

// ---------------------------------------------------------------------------
// K4: selected 128x128 img patch (with pad-32 zeros) -> bilinear 224x224.
// ---------------------------------------------------------------------------
__global__ void img_resize_kernel(const float* __restrict__ img,
                                  const int* __restrict__ sel,
                                  float* __restrict__ out) {
  const int total = BS * 3 * 224 * 224;
  int gid = blockIdx.x * blockDim.x + threadIdx.x;
  if (gid >= total) return;
  int xx = gid % 224;
  int yy = (gid / 224) % 224;
  int ch = (gid / (224 * 224)) % 3;
  int b  = gid / (3 * 224 * 224);
  int s = sel[b];
  int oy = (s / 9) * 48 - 32;   // patch origin in unpadded img coords
  int ox = (s % 9) * 48 - 32;
  const float sc = 128.0f / 224.0f;
  float sy = (yy + 0.5f) * sc - 0.5f;
  float sx = (xx + 0.5f) * sc - 0.5f;
  int iy0 = (int)floorf(sy); float fy = sy - (float)iy0;
  int ix0 = (int)floorf(sx); float fx = sx - (float)ix0;
  int iy0c = min(max(iy0, 0), 127), iy1c = min(max(iy0 + 1, 0), 127);
  int ix0c = min(max(ix0, 0), 127), ix1c = min(max(ix0 + 1, 0), 127);
  const float* ib = img + ((size_t)b * 3 + ch) * 448 * 448;
  auto samp = [&](int iy, int ix) -> float {
    int gy = oy + iy, gx = ox + ix;
    return ((unsigned)gy < 448u && (unsigned)gx < 448u)
               ? ib[(size_t)gy * 448 + gx] : 0.0f;
  };
  float v = (1.0f - fy) * ((1.0f - fx) * samp(iy0c, ix0c) + fx * samp(iy0c, ix1c))
          +          fy * ((1.0f - fx) * samp(iy1c, ix0c) + fx * samp(iy1c, ix1c));
  out[gid] = v;
}

extern "C" void kernel_launch(void* const* d_in, const int* in_sizes, int n_in,
                              void* d_out, int out_size, void* d_ws, size_t ws_size,
                              hipStream_t stream) {
  (void)in_sizes; (void)n_in; (void)out_size; (void)ws_size;
  const float* x    = (const float*)d_in[0];
  const float* mask = (const float*)d_in[1];
  const float* img  = (const float*)d_in[2];
  const float* W_fc = (const float*)d_in[3];
  const float* b_fc = (const float*)d_in[4];
  const float* W_l  = (const float*)d_in[5];
  const float* b_l  = (const float*)d_in[6];
  const float* W_m  = (const float*)d_in[7];
  const float* b_m  = (const float*)d_in[8];
  const float* W_s  = (const float*)d_in[9];
  const float* b_s  = (const float*)d_in[10];
  float* out = (float*)d_out;

  // workspace layout (floats): Wtp_fc | Wtp_parts | bias_cat | ml | sel
  float* ws        = (float*)d_ws;
  float2* Wtp_fc    = (float2*)ws;                         // (768/2)*208 float2
  float2* Wtp_parts = Wtp_fc + (C_DIM / 2) * NFC_PAD;      // (768/2)*768 float2
  float*  bias_cat  = (float*)(Wtp_parts + (C_DIM / 2) * C_DIM);  // 768
  float*  ml        = bias_cat + C_DIM;                    // 16*784
  int*    sel       = (int*)(ml + BS * 784);               // 16

  const int prepN = (C_DIM / 2) * NFC_PAD + (C_DIM / 2) * C_DIM + C_DIM;
  prep_kernel<<<(prepN + 255) / 256, 256, 0, stream>>>(
      W_fc, W_l, W_m, W_s, b_l, b_m, b_s, Wtp_fc, Wtp_parts, bias_cat);

  cls_softmax_kernel<<<784, 32, 0, stream>>>(x, Wtp_fc, b_fc, ml);

  select_kernel<<<BS, 128, 0, stream>>>(ml, mask, sel);

  parts_gemm_kernel<<<192, 32, 0, stream>>>(x, sel, Wtp_parts, bias_cat, out);

  const int imgN = BS * 3 * 224 * 224;
  img_resize_kernel<<<(imgN + 255) / 256, 256, 0, stream>>>(
      img, sel, out + (size_t)BS * 64 * C_DIM);
}